// BlockDiagonalGRU_32976758899294
// MI455X (gfx1250) — compile-verified
//
#include <hip/hip_runtime.h>
#include <hip/hip_bf16.h>
#include <math.h>

// Problem constants (match reference)
#define BATCH 8192
#define IN    1024
#define HDIM  2048
#define NB    8
#define BS    256           // hidden block size
#define M_T   64            // rows per workgroup tile
#define KC    16            // K-chunk staged in LDS (double buffered)
#define LDSP  (KC + 4)      // stride 20 words = 80B: 16B-aligned, bank-conflict-free

typedef float v2f __attribute__((ext_vector_type(2)));
typedef float v8f __attribute__((ext_vector_type(8)));

__device__ __forceinline__ float sigmoid_f(float v) {
    return 1.0f / (1.0f + __expf(-v));
}

// CDNA5 async LDS-DMA: copy 16 bytes global -> LDS per lane, ASYNCcnt-tracked.
__device__ __forceinline__ void async_b128(void* lds_ptr, const float* gaddr) {
    const unsigned loff = (unsigned)(uintptr_t)lds_ptr;   // low 32 bits = LDS offset
    asm volatile("global_load_async_to_lds_b128 %0, %1, off"
                 :: "v"(loff), "v"(gaddr) : "memory");
}
__device__ __forceinline__ void wait_async0() {
    asm volatile("s_wait_asynccnt 0x0" ::: "memory");
}

// Stage one KC-chunk of A (64 x KC) and B (256 x KC) into an LDS buffer pair
// using 5 async B128 DMAs per thread (no VGPR data staging).
__device__ __forceinline__ void stage_chunk(
    const float* __restrict__ A, int lda,
    const float* __restrict__ Bw, int ldb, int k0,
    float (*sA)[LDSP], float (*sB)[LDSP], int tid)
{
    {   // A tile: 64 rows x 16 floats = 256 B128 transfers -> 1 per thread
        const int r = tid >> 2;
        const int q = (tid & 3) * 4;
        async_b128(&sA[r][q], A + (size_t)r * lda + k0 + q);
    }
    {   // B tile: 256 rows x 16 floats -> 4 per thread (row = tid)
        const float* grow = Bw + (size_t)tid * ldb + k0;
#pragma unroll
        for (int j = 0; j < KC; j += 4)
            async_b128(&sB[tid][j], grow + j);
    }
}

// NT GEMM accumulate with double-buffered async staging:
// acc[t] += A[64 x K] * B[256 x K]^T on this wave's 16x128 slice (8 tiles).
// A,B are K-major, so A-frags and B-frags share identical K-major lane
// addressing for v_wmma_f32_16x16x4_f32.
__device__ __forceinline__ void mm_accum(
    const float* __restrict__ A, int lda,
    const float* __restrict__ Bw, int ldb,
    int K, v8f acc[8],
    float (*sA)[M_T][LDSP], float (*sB)[BS][LDSP],
    int tid, int rg, int ch, int half, int lrow)
{
    const int nch = K / KC;
    stage_chunk(A, lda, Bw, ldb, 0, sA[0], sB[0], tid);
    wait_async0();
    __syncthreads();
    for (int i = 0; i < nch; ++i) {
        const int cur = i & 1, nxt = cur ^ 1;
        if (i + 1 < nch)   // DMA next chunk while computing this one
            stage_chunk(A, lda, Bw, ldb, (i + 1) * KC, sA[nxt], sB[nxt], tid);
#pragma unroll
        for (int kk = 0; kk < KC; kk += 4) {
            v2f a;
            const int ar = 16 * rg + lrow;
            a.x = sA[cur][ar][kk + 2 * half];
            a.y = sA[cur][ar][kk + 2 * half + 1];
#pragma unroll
            for (int t = 0; t < 8; ++t) {
                const int n = 128 * ch + 16 * t + lrow;
                v2f b;
                b.x = sB[cur][n][kk + 2 * half];
                b.y = sB[cur][n][kk + 2 * half + 1];
                acc[t] = __builtin_amdgcn_wmma_f32_16x16x4_f32(
                    false, a, false, b, (short)0, acc[t], false, false);
            }
        }
        wait_async0();      // next chunk landed in LDS (this wave's DMAs)
        __syncthreads();    // ... and everyone else's
    }
}

__global__ __launch_bounds__(256)
void BlockDiagonalGRU_fused_kernel(
    const float* __restrict__ x,        // [B, IN]
    const float* __restrict__ h,        // [B, H]
    const float* __restrict__ W_ir,     // [3H, IN]
    const float* __restrict__ b_ir_lin, // [3H]
    const float* __restrict__ b_ir,     // [3H]
    const float* __restrict__ W_h,      // [NB, 3*BS, BS]
    const float* __restrict__ b_hr,     // [3H] (per-block local r,z,n layout)
    float* __restrict__ out)            // [B, H]
{
    __shared__ __align__(16) float sA[2][M_T][LDSP];  // 2 x 64 x 20  (10.2 KB)
    __shared__ __align__(16) float sB[2][BS][LDSP];   // 2 x 256 x 20 (40.9 KB)

    const int tid  = threadIdx.x;
    const int row0 = blockIdx.x * M_T;
    const int nb   = blockIdx.y;

    const int w    = tid >> 5;
    const int lane = tid & 31;
    const int rg   = w >> 1;     // row group: rows [16*rg, 16*rg+16), rg in 0..3
    const int ch   = w & 1;      // col half: cols [128*ch, 128*ch+128)
    const int half = lane >> 4;  // VGPR half select
    const int lrow = lane & 15;  // N (col) coordinate inside 16-wide tile

    const float* Ax = x + (size_t)row0 * IN;             // gemm1 A, lda=IN
    const float* Ah = h + (size_t)row0 * HDIM + nb * BS; // gemm2 A, lda=HDIM
    const size_t whblk = (size_t)nb * (3 * BS) * BS;

    v8f rv[8], nv[8], zv[8];

    // ---------------- gate r (g=0): shared accumulator for wx+wh -----------
    {
        v8f acc[8];
#pragma unroll
        for (int t = 0; t < 8; ++t)
#pragma unroll
            for (int v = 0; v < 8; ++v) acc[t][v] = 0.0f;

        mm_accum(Ax, IN, W_ir + (size_t)(0 * HDIM + nb * BS) * IN, IN,
                 IN, acc, sA, sB, tid, rg, ch, half, lrow);
        mm_accum(Ah, HDIM, W_h + whblk + (size_t)(0 * BS) * BS, BS,
                 BS, acc, sA, sB, tid, rg, ch, half, lrow);
#pragma unroll
        for (int t = 0; t < 8; ++t) {
            const int col = 128 * ch + 16 * t + lrow;
            const int gc  = 0 * HDIM + nb * BS + col;
            const float bias = b_ir_lin[gc] + b_ir[gc] + b_hr[nb * 3 * BS + 0 * BS + col];
#pragma unroll
            for (int v = 0; v < 8; ++v)
                rv[t][v] = sigmoid_f(acc[t][v] + bias);
        }
    }

    // ---------------- gate n (g=2): separate wx / wh accumulators ----------
    {
        v8f accx[8], acch[8];
#pragma unroll
        for (int t = 0; t < 8; ++t)
#pragma unroll
            for (int v = 0; v < 8; ++v) { accx[t][v] = 0.0f; acch[t][v] = 0.0f; }

        mm_accum(Ax, IN, W_ir + (size_t)(2 * HDIM + nb * BS) * IN, IN,
                 IN, accx, sA, sB, tid, rg, ch, half, lrow);
        mm_accum(Ah, HDIM, W_h + whblk + (size_t)(2 * BS) * BS, BS,
                 BS, acch, sA, sB, tid, rg, ch, half, lrow);
#pragma unroll
        for (int t = 0; t < 8; ++t) {
            const int col = 128 * ch + 16 * t + lrow;
            const int gc  = 2 * HDIM + nb * BS + col;
            const float bx = b_ir_lin[gc] + b_ir[gc];
            const float bh = b_hr[nb * 3 * BS + 2 * BS + col];
#pragma unroll
            for (int v = 0; v < 8; ++v)
                nv[t][v] = tanhf(accx[t][v] + bx + rv[t][v] * (acch[t][v] + bh));
        }
    }

    // ---------------- gate z (g=1): shared accumulator ---------------------
    {
        v8f acc[8];
#pragma unroll
        for (int t = 0; t < 8; ++t)
#pragma unroll
            for (int v = 0; v < 8; ++v) acc[t][v] = 0.0f;

        mm_accum(Ax, IN, W_ir + (size_t)(1 * HDIM + nb * BS) * IN, IN,
                 IN, acc, sA, sB, tid, rg, ch, half, lrow);
        mm_accum(Ah, HDIM, W_h + whblk + (size_t)(1 * BS) * BS, BS,
                 BS, acc, sA, sB, tid, rg, ch, half, lrow);
#pragma unroll
        for (int t = 0; t < 8; ++t) {
            const int col = 128 * ch + 16 * t + lrow;
            const int gc  = 1 * HDIM + nb * BS + col;
            const float bias = b_ir_lin[gc] + b_ir[gc] + b_hr[nb * 3 * BS + 1 * BS + col];
#pragma unroll
            for (int v = 0; v < 8; ++v)
                zv[t][v] = sigmoid_f(acc[t][v] + bias);
        }
    }

    // -------- combine + coalesced store, two 32-row passes through LDS -----
    float* sOut = &sB[0][0][0];      // reuse as [32][256] tile (32 KB < 40.9 KB)
#pragma unroll
    for (int p = 0; p < 2; ++p) {
        __syncthreads();             // LDS free (gemms done / prev pass stored)
        if ((rg >> 1) == p) {        // waves owning rows [32p, 32p+32)
#pragma unroll
            for (int t = 0; t < 8; ++t) {
                const int col = 128 * ch + 16 * t + lrow;
#pragma unroll
                for (int v = 0; v < 8; ++v) {
                    const int ml = 16 * (rg & 1) + v + 8 * half;   // row in pass
                    const int m  = 32 * p + ml;                    // row in tile
                    const float hb = h[(size_t)(row0 + m) * HDIM + nb * BS + col];
                    const float z = zv[t][v];
                    sOut[ml * BS + col] = (1.0f - z) * hb + z * nv[t][v];
                }
            }
        }
        __syncthreads();
        // 32 rows x 256 cols, one full row per iteration, fully coalesced
        for (int r = 0; r < 32; ++r)
            out[(size_t)(row0 + 32 * p + r) * HDIM + nb * BS + tid] =
                sOut[r * BS + tid];
    }
}

extern "C" void kernel_launch(void* const* d_in, const int* in_sizes, int n_in,
                              void* d_out, int out_size, void* d_ws, size_t ws_size,
                              hipStream_t stream) {
    (void)in_sizes; (void)n_in; (void)d_ws; (void)ws_size; (void)out_size;
    const float* x        = (const float*)d_in[0];
    const float* h        = (const float*)d_in[1];
    const float* W_ir     = (const float*)d_in[2];
    const float* b_ir_lin = (const float*)d_in[3];
    const float* b_ir     = (const float*)d_in[4];
    const float* W_h      = (const float*)d_in[5];
    const float* b_hr     = (const float*)d_in[6];
    float* out            = (float*)d_out;

    dim3 grid(BATCH / M_T, NB);   // 128 x 8 workgroups
    dim3 block(256);              // 8 wave32s
    BlockDiagonalGRU_fused_kernel<<<grid, block, 0, stream>>>(
        x, h, W_ir, b_ir_lin, b_ir, W_h, b_hr, out);
}